// GATClassifier_46892452938392
// MI455X (gfx1250) — compile-verified
//
#include <hip/hip_runtime.h>
#include <math.h>

// Problem constants (match reference)
#define F_IN      128
#define HID       64
#define H1        8
#define C1        8
#define H2        1
#define C2        64
#define G_GRAPHS  128
#define NUM_CLASS 10
#define NEG_SLOPE 0.2f

typedef __attribute__((ext_vector_type(2))) float v2f;
typedef __attribute__((ext_vector_type(8))) float v8f;

// ---------------- utility ----------------

__device__ __forceinline__ float atomicMaxFloat(float* addr, float value) {
  // IEEE-754 trick: signed-int max for non-negative, unsigned min for negative.
  // Works when buffer initialized to -inf (0xFF800000).
  if (value >= 0.0f)
    return __int_as_float(atomicMax((int*)addr, __float_as_int(value)));
  else
    return __uint_as_float(atomicMin((unsigned int*)addr, __float_as_uint(value)));
}

__global__ void fill_kernel(float* __restrict__ p, float v, int n) {
  int i = blockIdx.x * blockDim.x + threadIdx.x;
  if (i < n) p[i] = v;
}

// ---------------- WMMA GEMM: Y[nrows,64] = X[nrows,K] @ W[K,64] ----------------
// One wave32 per 16-row tile; 4 N-tiles of 16; K-loop step 4 using
// V_WMMA_F32_16X16X4_F32.
// W is staged in LDS PRE-PAIRED: Wlds[(k/2)*64 + n] = {W[k][n], W[k+1][n]}
// so each B fragment is one aligned ds_load_b64 straight into an even VGPR
// pair (no repacking movs). A fragment is one aligned global b64 load.
template<int K>
__global__ void gemm_wmma_kernel(const float* __restrict__ X,
                                 const float* __restrict__ W,
                                 float* __restrict__ Y, int nrows) {
  __shared__ v2f Wlds[(K / 2) * HID];
  for (int i = threadIdx.x; i < (K / 2) * HID; i += blockDim.x) {
    int kk = i / HID, n = i % HID;
    v2f t;
    t.x = W[(2 * kk) * HID + n];
    t.y = W[(2 * kk + 1) * HID + n];
    Wlds[i] = t;
  }
  __syncthreads();

  const int wave = threadIdx.x >> 5;
  const int lane = threadIdx.x & 31;
  const int tile = blockIdx.x * (blockDim.x >> 5) + wave;
  if (tile * 16 >= nrows) return;               // wave-uniform: EXEC stays all-1s

  const int row0  = tile * 16;
  const int mrow  = lane & 15;                  // A: M index
  const int khalf = (lane >> 4) * 2;            // A/B: lanes 16-31 hold K=2,3
  const int ncol  = lane & 15;                  // B: N index within tile
  const float* __restrict__ xrow = X + (size_t)(row0 + mrow) * K;

  v8f acc0 = {}, acc1 = {}, acc2 = {}, acc3 = {};
  for (int k = 0; k < K; k += 4) {
    v2f a = *(const v2f*)(xrow + k + khalf);    // {X[m][k+kh], X[m][k+kh+1]}
    const v2f* __restrict__ wrow = &Wlds[((k + khalf) >> 1) * HID];
    v2f b0 = wrow[ncol];
    v2f b1 = wrow[16 + ncol];
    v2f b2 = wrow[32 + ncol];
    v2f b3 = wrow[48 + ncol];
    acc0 = __builtin_amdgcn_wmma_f32_16x16x4_f32(false, a, false, b0, (short)0, acc0, false, false);
    acc1 = __builtin_amdgcn_wmma_f32_16x16x4_f32(false, a, false, b1, (short)0, acc1, false, false);
    acc2 = __builtin_amdgcn_wmma_f32_16x16x4_f32(false, a, false, b2, (short)0, acc2, false, false);
    acc3 = __builtin_amdgcn_wmma_f32_16x16x4_f32(false, a, false, b3, (short)0, acc3, false, false);
  }

  // C/D layout: lanes 0-15 -> rows M=0..7 in v[0..7], lanes 16-31 -> rows M=8..15
  const int col   = lane & 15;
  const int mbase = (lane >> 4) * 8;
#pragma unroll
  for (int m = 0; m < 8; ++m) {
    float* y = Y + (size_t)(row0 + mbase + m) * HID;
    y[col]      = acc0[m];
    y[16 + col] = acc1[m];
    y[32 + col] = acc2[m];
    y[48 + col] = acc3[m];
  }
}

// ---------------- attention coefficients e_src/e_dst per node ----------------
template<int H, int C>
__global__ void attn_coef_kernel(const float* __restrict__ xh,
                                 const float* __restrict__ a_src,
                                 const float* __restrict__ a_dst,
                                 float* __restrict__ es, float* __restrict__ ed,
                                 int n) {
  int i = blockIdx.x * blockDim.x + threadIdx.x;
  if (i >= n) return;
  const float* row = xh + (size_t)i * (H * C);
#pragma unroll
  for (int h = 0; h < H; ++h) {
    float s = 0.f, d = 0.f;
#pragma unroll
    for (int c = 0; c < C; ++c) {
      float v = row[h * C + c];
      s += v * a_src[h * C + c];
      d += v * a_dst[h * C + c];
    }
    es[i * H + h] = s;
    ed[i * H + h] = d;
  }
}

// ---------------- edge pass 1: segment max ----------------
template<int H>
__global__ void edge_max_kernel(const int* __restrict__ src, const int* __restrict__ dst,
                                const float* __restrict__ es, const float* __restrict__ ed,
                                float* __restrict__ m, int total /* = E*H */) {
  int t = blockIdx.x * blockDim.x + threadIdx.x;
  if (t >= total) return;
  int e = t / H, h = t % H;
  int s = src[e], d = dst[e];
  float v = es[s * H + h] + ed[d * H + h];
  v = v > 0.f ? v : NEG_SLOPE * v;
  atomicMaxFloat(&m[d * H + h], v);
}

// ---------------- edge pass 2: segment sum of exp ----------------
template<int H>
__global__ void edge_sum_kernel(const int* __restrict__ src, const int* __restrict__ dst,
                                const float* __restrict__ es, const float* __restrict__ ed,
                                const float* __restrict__ m, float* __restrict__ ssum,
                                int total /* = E*H */) {
  int t = blockIdx.x * blockDim.x + threadIdx.x;
  if (t >= total) return;
  int e = t / H, h = t % H;
  int s = src[e], d = dst[e];
  float v = es[s * H + h] + ed[d * H + h];
  v = v > 0.f ? v : NEG_SLOPE * v;
  float ex = expf(v - m[d * H + h]);
  atomicAdd(&ssum[d * H + h], ex);
}

// ---------------- edge pass 3: alpha + weighted aggregation ----------------
// Thread per (edge, head, 8-channel chunk); writes alpha for chunk 0 if requested.
template<int H, int C>
__global__ void edge_aggregate_kernel(const int* __restrict__ src, const int* __restrict__ dst,
                                      const float* __restrict__ es, const float* __restrict__ ed,
                                      const float* __restrict__ m, const float* __restrict__ ssum,
                                      const float* __restrict__ xh,
                                      float* __restrict__ agg,
                                      float* __restrict__ alphaOut,
                                      int total /* = E*H*(C/8) */) {
  int t = blockIdx.x * blockDim.x + threadIdx.x;
  if (t >= total) return;
  const int chunks = C / 8;
  int e  = t / (H * chunks);
  int r  = t % (H * chunks);
  int h  = r / chunks;
  int c0 = (r % chunks) * 8;
  int s = src[e], d = dst[e];
  float v = es[s * H + h] + ed[d * H + h];
  v = v > 0.f ? v : NEG_SLOPE * v;
  float ex    = expf(v - m[d * H + h]);
  float alpha = ex / (ssum[d * H + h] + 1e-16f);
  if (alphaOut != nullptr && c0 == 0) alphaOut[e * H + h] = alpha;
  const float* xs = xh + (size_t)s * (H * C) + h * C + c0;
  float* ad       = agg + (size_t)d * (H * C) + h * C + c0;
#pragma unroll
  for (int c = 0; c < 8; ++c) atomicAdd(&ad[c], alpha * xs[c]);
}

// ---------------- bias (+ optional ReLU), in place ----------------
__global__ void bias_act_kernel(float* __restrict__ h, const float* __restrict__ b,
                                int total, int relu) {
  int t = blockIdx.x * blockDim.x + threadIdx.x;
  if (t >= total) return;
  float v = h[t] + b[t & (HID - 1)];
  h[t] = relu ? fmaxf(v, 0.f) : v;
}

// ---------------- global mean-pool accumulation ----------------
__global__ void pool_kernel(const float* __restrict__ h, const int* __restrict__ batch,
                            float* __restrict__ pool, float* __restrict__ cnt, int n) {
  int t = blockIdx.x * blockDim.x + threadIdx.x;
  if (t >= n * HID) return;
  int i = t >> 6, c = t & 63;
  int b = batch[i];
  atomicAdd(&pool[b * HID + c], h[(size_t)i * HID + c]);
  if (c == 0) atomicAdd(&cnt[b], 1.0f);
}

// ---------------- final MLP: 64 -> 32 -> 16 -> 10 ----------------
__global__ void mlp_kernel(const float* __restrict__ pool, const float* __restrict__ cnt,
                           const float* __restrict__ wA, const float* __restrict__ bA,
                           const float* __restrict__ wB, const float* __restrict__ bB,
                           const float* __restrict__ wC, const float* __restrict__ bC,
                           float* __restrict__ logits) {
  int g = blockIdx.x * blockDim.x + threadIdx.x;
  if (g >= G_GRAPHS) return;
  float inv = 1.0f / fmaxf(cnt[g], 1.0f);
  float p[HID];
#pragma unroll
  for (int c = 0; c < HID; ++c) p[c] = pool[g * HID + c] * inv;
  float zA[32];
  for (int j = 0; j < 32; ++j) {
    float acc = bA[j];
    for (int c = 0; c < HID; ++c) acc += p[c] * wA[c * 32 + j];
    zA[j] = fmaxf(acc, 0.f);
  }
  float zB[16];
  for (int j = 0; j < 16; ++j) {
    float acc = bB[j];
    for (int c = 0; c < 32; ++c) acc += zA[c] * wB[c * 16 + j];
    zB[j] = fmaxf(acc, 0.f);
  }
  for (int j = 0; j < NUM_CLASS; ++j) {
    float acc = bC[j];
    for (int c = 0; c < 16; ++c) acc += zB[c] * wC[c * 10 + j];
    logits[g * NUM_CLASS + j] = acc;
  }
}

// ---------------- launcher ----------------
extern "C" void kernel_launch(void* const* d_in, const int* in_sizes, int n_in,
                              void* d_out, int out_size, void* d_ws, size_t ws_size,
                              hipStream_t stream) {
  const float* x      = (const float*)d_in[0];
  const int*   eidx   = (const int*)d_in[1];
  const int*   batch  = (const int*)d_in[2];
  const float* w1     = (const float*)d_in[3];
  const float* a_src1 = (const float*)d_in[4];
  const float* a_dst1 = (const float*)d_in[5];
  const float* b1     = (const float*)d_in[6];
  const float* w2     = (const float*)d_in[7];
  const float* a_src2 = (const float*)d_in[8];
  const float* a_dst2 = (const float*)d_in[9];
  const float* b2     = (const float*)d_in[10];
  const float* wA     = (const float*)d_in[11];
  const float* bA     = (const float*)d_in[12];
  const float* wB     = (const float*)d_in[13];
  const float* bB     = (const float*)d_in[14];
  const float* wC     = (const float*)d_in[15];
  const float* bC     = (const float*)d_in[16];

  const int N = in_sizes[0] / F_IN;     // 50000
  const int E = in_sizes[1] / 2;        // 1600000
  const int* src = eidx;
  const int* dst = eidx + E;

  // ---- workspace layout (floats) ----
  float* ws = (float*)d_ws;
  float* xh1  = ws;                 ws += (size_t)N * HID;   // conv1 x@W1
  float* xh2  = ws;                 ws += (size_t)N * HID;   // conv2 h1@W2
  float* es1  = ws;                 ws += (size_t)N * H1;
  float* ed1  = ws;                 ws += (size_t)N * H1;
  float* es2  = ws;                 ws += (size_t)N * H2;
  float* ed2  = ws;                 ws += (size_t)N * H2;
  // zero-init region (contiguous)
  float* zero0 = ws;
  float* agg1 = ws;                 ws += (size_t)N * HID;
  float* agg2 = ws;                 ws += (size_t)N * HID;
  float* s1   = ws;                 ws += (size_t)N * H1;
  float* s2   = ws;                 ws += (size_t)N * H2;
  float* pool = ws;                 ws += (size_t)G_GRAPHS * HID;
  float* cnt  = ws;                 ws += (size_t)G_GRAPHS;
  int zeroN = (int)(ws - zero0);
  // -inf init region (contiguous)
  float* ninf0 = ws;
  float* m1   = ws;                 ws += (size_t)N * H1;
  float* m2   = ws;                 ws += (size_t)N * H2;
  int ninfN = (int)(ws - ninf0);

  float* logits_out = (float*)d_out;                         // [G, 10]
  float* alpha_out  = (float*)d_out + G_GRAPHS * NUM_CLASS;  // [E, 1]

  const int BT = 256;
  // init
  fill_kernel<<<(zeroN + BT - 1) / BT, BT, 0, stream>>>(zero0, 0.0f, zeroN);
  fill_kernel<<<(ninfN + BT - 1) / BT, BT, 0, stream>>>(ninf0, -INFINITY, ninfN);

  // ---- conv1 ----
  int tiles = (N + 15) / 16;
  int wpb = BT / 32;
  gemm_wmma_kernel<F_IN><<<(tiles + wpb - 1) / wpb, BT, 0, stream>>>(x, w1, xh1, N);
  attn_coef_kernel<H1, C1><<<(N + BT - 1) / BT, BT, 0, stream>>>(xh1, a_src1, a_dst1, es1, ed1, N);
  {
    int tot = E * H1;
    edge_max_kernel<H1><<<(tot + BT - 1) / BT, BT, 0, stream>>>(src, dst, es1, ed1, m1, tot);
    edge_sum_kernel<H1><<<(tot + BT - 1) / BT, BT, 0, stream>>>(src, dst, es1, ed1, m1, s1, tot);
    int tot3 = E * H1 * (C1 / 8);
    edge_aggregate_kernel<H1, C1><<<(tot3 + BT - 1) / BT, BT, 0, stream>>>(
        src, dst, es1, ed1, m1, s1, xh1, agg1, nullptr, tot3);
  }
  bias_act_kernel<<<((size_t)N * HID + BT - 1) / BT, BT, 0, stream>>>(agg1, b1, N * HID, 1);

  // ---- conv2 ----
  gemm_wmma_kernel<HID><<<(tiles + wpb - 1) / wpb, BT, 0, stream>>>(agg1, w2, xh2, N);
  attn_coef_kernel<H2, C2><<<(N + BT - 1) / BT, BT, 0, stream>>>(xh2, a_src2, a_dst2, es2, ed2, N);
  {
    int tot = E * H2;
    edge_max_kernel<H2><<<(tot + BT - 1) / BT, BT, 0, stream>>>(src, dst, es2, ed2, m2, tot);
    edge_sum_kernel<H2><<<(tot + BT - 1) / BT, BT, 0, stream>>>(src, dst, es2, ed2, m2, s2, tot);
    int tot3 = E * H2 * (C2 / 8);
    edge_aggregate_kernel<H2, C2><<<(tot3 + BT - 1) / BT, BT, 0, stream>>>(
        src, dst, es2, ed2, m2, s2, xh2, agg2, alpha_out, tot3);
  }
  bias_act_kernel<<<((size_t)N * HID + BT - 1) / BT, BT, 0, stream>>>(agg2, b2, N * HID, 0);

  // ---- pool + MLP ----
  pool_kernel<<<((size_t)N * HID + BT - 1) / BT, BT, 0, stream>>>(agg2, batch, pool, cnt, N);
  mlp_kernel<<<1, G_GRAPHS, 0, stream>>>(pool, cnt, wA, bA, wB, bB, wC, bC, logits_out);
}